// LAN_36730560315737
// MI455X (gfx1250) — compile-verified
//
#include <hip/hip_runtime.h>
#include <hip/hip_bf16.h>
#include <math.h>

typedef __attribute__((ext_vector_type(2))) float v2f;
typedef __attribute__((ext_vector_type(4))) float v4f;
typedef __attribute__((ext_vector_type(8))) float v8f;

// Problem constants
#define NN 8
#define DD 4
#define CC 64
#define HW 25600              // 160*160
#define KK 1638400            // C*H*W
#define K4 409600             // KK/4 (float4 elements)
#define DC 256                // D*C
#define CHUNKS 256            // gram chunks per n

// Workspace layout (floats)
#define SCORES_OFF 0          // 128
#define L_OFF      128        // 128
#define WEFF_OFF   256        // 8*256*64 = 131072  (paired: W2[n][k/2][c][2])
#define PART_OFF   131328     // 8*256*10 = 20480

// ---------------------------------------------------------------------------
// Kernel 1: per-chunk partial Gram matrices.  x viewed as (N, 4, K).
// float4 (b128) loads; 10 symmetric v4f accumulators -> packed f32 FMAs.
// ---------------------------------------------------------------------------
__global__ __launch_bounds__(256) void LAN_gram_partial(
    const float* __restrict__ x, float* __restrict__ part) {
  const int n     = blockIdx.x >> 8;     // 256 chunks per n
  const int chunk = blockIdx.x & 255;
  const v4f* xn = (const v4f*)(x + (size_t)n * DD * KK);

  v4f s[10];
#pragma unroll
  for (int i = 0; i < 10; ++i) s[i] = (v4f){};

  for (int k = chunk * 256 + threadIdx.x; k < K4; k += CHUNKS * 256) {
    v4f v0 = xn[k];
    v4f v1 = xn[K4 + k];
    v4f v2 = xn[2 * K4 + k];
    v4f v3 = xn[3 * K4 + k];
    s[0] += v0 * v0; s[1] += v0 * v1; s[2] += v0 * v2; s[3] += v0 * v3;
    s[4] += v1 * v1; s[5] += v1 * v2; s[6] += v1 * v3;
    s[7] += v2 * v2; s[8] += v2 * v3;
    s[9] += v3 * v3;
  }

  float h[10];
#pragma unroll
  for (int i = 0; i < 10; ++i) h[i] = (s[i].x + s[i].y) + (s[i].z + s[i].w);

  // wave32 butterfly reduce (warpSize == 32 on gfx1250)
#pragma unroll
  for (int i = 0; i < 10; ++i)
#pragma unroll
    for (int off = 16; off > 0; off >>= 1) h[i] += __shfl_xor(h[i], off, 32);

  __shared__ float red[8][10];
  const int wave = threadIdx.x >> 5;
  const int lane = threadIdx.x & 31;
  if (lane == 0) {
#pragma unroll
    for (int i = 0; i < 10; ++i) red[wave][i] = h[i];
  }
  __syncthreads();
  if (threadIdx.x < 10) {
    float t = 0.0f;
#pragma unroll
    for (int w = 0; w < 8; ++w) t += red[w][threadIdx.x];
    part[(size_t)blockIdx.x * 10 + threadIdx.x] = t;
  }
}

// ---------------------------------------------------------------------------
// Kernel 2: deterministic final reduction; expand symmetric 10 -> full 16.
// ---------------------------------------------------------------------------
__global__ void LAN_gram_reduce(const float* __restrict__ part,
                                float* __restrict__ scores) {
  const int t = threadIdx.x;  // 128 threads: (n, d*4+e)
  if (t < 128) {
    const int n = t >> 4;
    int d = (t >> 2) & 3, e = t & 3;
    if (d > e) { int tmp = d; d = e; e = tmp; }
    const int pidx = d * 4 + e - (d * (d + 1)) / 2;  // symmetric pair index
    float s = 0.0f;
    for (int c = 0; c < CHUNKS; ++c)
      s += part[((size_t)(n * CHUNKS + c)) * 10 + pidx];
    scores[t] = s;
  }
}

// ---------------------------------------------------------------------------
// Kernel 3: softmax over the batch axis n for each (d,e).
// ---------------------------------------------------------------------------
__global__ void LAN_softmax_n(const float* __restrict__ scores,
                              float* __restrict__ L) {
  const int t = threadIdx.x;  // 16 threads, one (d,e) each
  if (t < 16) {
    float v[NN];
    float m = -INFINITY;
#pragma unroll
    for (int n = 0; n < NN; ++n) {
      v[n] = scores[n * 16 + t];
      m = fmaxf(m, v[n]);
    }
    float s = 0.0f;
#pragma unroll
    for (int n = 0; n < NN; ++n) {
      v[n] = __expf(v[n] - m);
      s += v[n];
    }
    const float inv = 1.0f / s;
#pragma unroll
    for (int n = 0; n < NN; ++n) L[n * 16 + t] = v[n] * inv;
  }
}

// ---------------------------------------------------------------------------
// Kernel 4: fold correlation into the conv weight.
// Weff[n][c][k] = W[c][k] + sum_d L[n][d][e] * W[c][d*64+ci], k = e*64+ci.
// Stored with consecutive-K pairs interleaved per channel:
//   element (n,k,c) at  n*16384 + (k>>1)*128 + c*2 + (k&1)
// so the GEMM A-fragment is a single aligned b64 load.
// ---------------------------------------------------------------------------
__global__ __launch_bounds__(256) void LAN_make_weff(
    const float* __restrict__ w, const float* __restrict__ L,
    float* __restrict__ wt) {
  const int idx = blockIdx.x * 256 + threadIdx.x;  // 131072 total
  const int c  = idx & 63;
  const int k  = (idx >> 6) & 255;
  const int n  = idx >> 14;
  const int e  = k >> 6;
  const int ci = k & 63;
  float acc = w[c * DC + k];
#pragma unroll
  for (int d = 0; d < 4; ++d)
    acc += L[n * 16 + d * 4 + e] * w[c * DC + d * 64 + ci];
  wt[(size_t)n * 16384 + (k >> 1) * 128 + c * 2 + (k & 1)] = acc;
}

// ---------------------------------------------------------------------------
// Kernel 5: per-n GEMM  out[n] (64 x 25600) = Weff[n] (64x256) @ X[n]
// (256 x 25600) + bias, via V_WMMA_F32_16X16X4_F32.
// 8 waves/block; each wave owns a 64(M) x 32(N) strip: 8 v8f accumulators,
// A fragments (b64 loads) reused across both N-tiles -> 8 vmem : 8 wmma.
// ---------------------------------------------------------------------------
__global__ __launch_bounds__(256) void LAN_gemm_wmma(
    const float* __restrict__ x, const float* __restrict__ wt,
    const float* __restrict__ bias, float* __restrict__ out) {
  const int wave = threadIdx.x >> 5;
  const int lane = threadIdx.x & 31;
  const int half = lane >> 4;   // 0: K {0,1} / M r ; 1: K {2,3} / M r+8
  const int col  = lane & 15;

  const int n  = blockIdx.x / 100;      // 100 blocks per n (256 cols each)
  const int pb = blockIdx.x % 100;
  const int p0 = pb * 256 + wave * 32;

  const float* Xn = x   + (size_t)n * DC * HW;   // (256, 25600) row-major
  const v2f*   Wp = (const v2f*)(wt + (size_t)n * DC * CC);  // paired layout
  float*       On = out + (size_t)n * CC * HW;   // (64, 25600)  row-major

  v8f acc[2][4];
#pragma unroll
  for (int t = 0; t < 2; ++t)
#pragma unroll
    for (int m = 0; m < 4; ++m) acc[t][m] = (v8f){};

  for (int kk = 0; kk < 64; ++kk) {
    const int kb = kk * 4 + half * 2;  // even

    // Two B fragments: 4x16 tiles of X (each load: two full 64B lines)
    const float* xr = Xn + (size_t)kb * HW + p0 + col;
    v2f b0, b1;
    b0.x = xr[0];   b0.y = xr[HW];
    b1.x = xr[16];  b1.y = xr[HW + 16];

    const v2f* wrow = Wp + (size_t)(kb >> 1) * 64;  // 64 v2f per K-pair row
#pragma unroll
    for (int m = 0; m < 4; ++m) {
      const v2f a = wrow[m * 16 + col];  // single b64 A-fragment load
      acc[0][m] = __builtin_amdgcn_wmma_f32_16x16x4_f32(
          false, a, false, b0, (short)0, acc[0][m], false, false);
      acc[1][m] = __builtin_amdgcn_wmma_f32_16x16x4_f32(
          false, a, false, b1, (short)0, acc[1][m], false, false);
    }
  }

  // Store: VGPR r holds M = r (lanes 0-15) / M = r+8 (lanes 16-31)
#pragma unroll
  for (int t = 0; t < 2; ++t)
#pragma unroll
    for (int m = 0; m < 4; ++m)
#pragma unroll
      for (int r = 0; r < 8; ++r) {
        const int row = m * 16 + half * 8 + r;  // output channel c
        On[(size_t)row * HW + p0 + t * 16 + col] = acc[t][m][r] + bias[row];
      }
}

extern "C" void kernel_launch(void* const* d_in, const int* in_sizes, int n_in,
                              void* d_out, int out_size, void* d_ws,
                              size_t ws_size, hipStream_t stream) {
  const float* x      = (const float*)d_in[0];
  const float* conv_w = (const float*)d_in[1];
  const float* conv_b = (const float*)d_in[2];
  float* out = (float*)d_out;
  float* ws  = (float*)d_ws;

  float* scores = ws + SCORES_OFF;
  float* L      = ws + L_OFF;
  float* weff   = ws + WEFF_OFF;
  float* part   = ws + PART_OFF;

  LAN_gram_partial<<<NN * CHUNKS, 256, 0, stream>>>(x, part);
  LAN_gram_reduce<<<1, 128, 0, stream>>>(part, scores);
  LAN_softmax_n<<<1, 16, 0, stream>>>(scores, L);
  LAN_make_weff<<<(NN * DC * CC) / 256, 256, 0, stream>>>(conv_w, L, weff);
  LAN_gemm_wmma<<<NN * 100, 256, 0, stream>>>(x, weff, conv_b, out);
}